// SetConv_18588618457255
// MI455X (gfx1250) — compile-verified
//
#include <hip/hip_runtime.h>
#include <math.h>

#define BN_EPS 0.001f
static constexpr int NPTS  = 4096;
static constexpr int BATCH = 4;
static constexpr int KNN   = 32;
static constexpr int M_TOT = BATCH * NPTS * KNN;   // 524288
static constexpr int OUT_PTS = BATCH * 3 * NPTS;   // 49152

typedef float v2f __attribute__((ext_vector_type(2)));
typedef float v8f __attribute__((ext_vector_type(8)));

// -------------------- new_points = points (fps_idx == arange(3)) ---------
__global__ __launch_bounds__(256)
void copy_points_kernel(const float* __restrict__ pts, float* __restrict__ out) {
    int i = blockIdx.x * 256 + threadIdx.x;
    if (i < OUT_PTS) out[i] = pts[i];
}

// -------------------- kNN (top-32 by pd = -||xi-xj||^2) ------------------
__global__ __launch_bounds__(256)
void knn_kernel(const float* __restrict__ pts, int* __restrict__ idx) {
    __shared__ float X[NPTS], Y[NPTS], Z[NPTS], XX[NPTS];   // 64 KB
    __shared__ float lval[KNN * 256];                        // 32 KB
    __shared__ int   lind[KNN * 256];                        // 32 KB
    const int tid = threadIdx.x;
    const int b   = blockIdx.y;
    const float* pb = pts + (size_t)b * 3 * NPTS;
    for (int i = tid; i < NPTS; i += 256) {
        float x = pb[i], y = pb[NPTS + i], z = pb[2 * NPTS + i];
        X[i] = x; Y[i] = y; Z[i] = z; XX[i] = x * x + y * y + z * z;
    }
    #pragma unroll
    for (int j = 0; j < KNN; ++j) { lval[j * 256 + tid] = -1e30f; lind[j * 256 + tid] = 0; }
    __syncthreads();
    const int n = blockIdx.x * 256 + tid;
    const float qx = X[n], qy = Y[n], qz = Z[n], qq = XX[n];
    float vmin = -1e30f;
    for (int m = 0; m < NPTS; ++m) {
        float dot = qx * X[m] + qy * Y[m] + qz * Z[m];
        float pd  = (-qq) - (-2.0f * dot) - XX[m];           // match reference fp order
        if (pd > vmin) {                                     // strict: keeps earlier index on ties
            int pos = KNN - 1;
            while (pos > 0 && lval[(pos - 1) * 256 + tid] < pd) {
                lval[pos * 256 + tid] = lval[(pos - 1) * 256 + tid];
                lind[pos * 256 + tid] = lind[(pos - 1) * 256 + tid];
                --pos;
            }
            lval[pos * 256 + tid] = pd;
            lind[pos * 256 + tid] = m;
            vmin = lval[(KNN - 1) * 256 + tid];
        }
    }
    int* op = idx + ((size_t)(b * NPTS + n)) * KNN;
    for (int j = 0; j < KNN; ++j) op[j] = lind[j * 256 + tid];
}

// -------------------- layer1: gather + 6->64 conv (stores raw y1) --------
__global__ __launch_bounds__(256)
void layer1_kernel(const float* __restrict__ pts, const int* __restrict__ idx,
                   const float* __restrict__ W0, const float* __restrict__ b0,
                   float* __restrict__ Yout) {
    __shared__ float sW[64 * 6];
    __shared__ float sb[64];
    const int tid = threadIdx.x;
    for (int i = tid; i < 64 * 6; i += 256) sW[i] = W0[i];
    if (tid < 64) sb[tid] = b0[tid];
    __syncthreads();
    const int m = blockIdx.x * 256 + tid;
    const int p = m >> 5;
    const int b = p >> 12;
    const int n = p & (NPTS - 1);
    const int j = idx[m];
    const float* pb = pts + (size_t)b * 3 * NPTS;
    const float cx = pb[n],       cy = pb[NPTS + n],       cz = pb[2 * NPTS + n];
    const float ex = pb[j] - cx,  ey = pb[NPTS + j] - cy,  ez = pb[2 * NPTS + j] - cz;
    #pragma unroll 8
    for (int o = 0; o < 64; ++o) {
        const float* w = &sW[o * 6];
        float y = sb[o] + w[0]*ex + w[1]*ey + w[2]*ez + w[3]*cx + w[4]*cy + w[5]*cz;
        Yout[(size_t)o * M_TOT + m] = y;
    }
}

// -------------------- per-channel partial sum/sumsq over stored rows -----
__global__ __launch_bounds__(256)
void colstats_kernel(const float* __restrict__ Y, float* __restrict__ psum,
                     float* __restrict__ psq, int NB) {
    const int c = blockIdx.y;
    const int chunk = M_TOT / NB;
    const float* row = Y + (size_t)c * M_TOT + (size_t)blockIdx.x * chunk;
    float s = 0.f, q = 0.f;
    for (int i = threadIdx.x; i < chunk; i += 256) { float v = row[i]; s += v; q += v * v; }
    __shared__ float ls[256], lq[256];
    ls[threadIdx.x] = s; lq[threadIdx.x] = q; __syncthreads();
    for (int st = 128; st > 0; st >>= 1) {
        if (threadIdx.x < st) { ls[threadIdx.x] += ls[threadIdx.x + st]; lq[threadIdx.x] += lq[threadIdx.x + st]; }
        __syncthreads();
    }
    if (threadIdx.x == 0) {
        psum[(size_t)c * NB + blockIdx.x] = ls[0];
        psq [(size_t)c * NB + blockIdx.x] = lq[0];
    }
}

// -------------------- reduce partials -> A = g*rsqrt(var+eps), B = be-A*mean
__global__ __launch_bounds__(256)
void finalize_kernel(const float* __restrict__ psum, const float* __restrict__ psq,
                     const float* __restrict__ g, const float* __restrict__ be,
                     float* __restrict__ As, float* __restrict__ Bs, int NB) {
    const int o = blockIdx.x;
    float s = 0.f, q = 0.f;
    for (int i = threadIdx.x; i < NB; i += 256) {
        s += psum[(size_t)o * NB + i];
        q += psq [(size_t)o * NB + i];
    }
    __shared__ float ls[256], lq[256];
    ls[threadIdx.x] = s; lq[threadIdx.x] = q; __syncthreads();
    for (int st = 128; st > 0; st >>= 1) {
        if (threadIdx.x < st) { ls[threadIdx.x] += ls[threadIdx.x + st]; lq[threadIdx.x] += lq[threadIdx.x + st]; }
        __syncthreads();
    }
    if (threadIdx.x == 0) {
        const float inv = 1.0f / (float)M_TOT;
        float mean = ls[0] * inv;
        float var  = lq[0] * inv - mean * mean;
        float A    = g[o] / sqrtf(var + BN_EPS);
        As[o] = A;
        Bs[o] = be[o] - A * mean;
    }
}

// -------------------- in-place f = relu(A*y + B) -------------------------
__global__ __launch_bounds__(256)
void normrelu_kernel(float* __restrict__ Y, const float* __restrict__ As,
                     const float* __restrict__ Bs) {
    const int c = blockIdx.y;
    const float A = As[c], Bv = Bs[c];
    size_t base = (size_t)c * M_TOT + (size_t)blockIdx.x * 1024 + (size_t)threadIdx.x * 4;
    float4* p = (float4*)(Y + base);
    float4 v = *p;
    v.x = fmaxf(A * v.x + Bv, 0.f);
    v.y = fmaxf(A * v.y + Bv, 0.f);
    v.z = fmaxf(A * v.z + Bv, 0.f);
    v.w = fmaxf(A * v.w + Bv, 0.f);
    *p = v;
}

// -------------------- WMMA GEMM: D[O x M] = W[O x C] * F[C x M] + bias ---
// MODE 0: store raw y + stat partials (layer 2)
// MODE 1: stat partials only           (layer 3, pass A)
// MODE 2: relu(A*y+B), max over K=32, write d_out (layer 3, pass B)
// Block: 256 threads = 8 waves; wave w owns 16 columns; block covers 128 m
// (= 4 points x 32 neighbors). All O/16 o-tiles accumulated per wave so F is
// streamed exactly once per pass; 8 waves share one weight-LDS image for
// better waves/SIMD than a 128-thread block at the same LDS footprint.
template <int C, int O, int MODE>
__global__ __launch_bounds__(256)
void gemm_wmma_kernel(const float* __restrict__ F, const float* __restrict__ W,
                      const float* __restrict__ bias, float* __restrict__ Yout,
                      float* __restrict__ psum, float* __restrict__ psq,
                      const float* __restrict__ As, const float* __restrict__ Bs,
                      float* __restrict__ out, int NB) {
    constexpr int OT = O / 16;
    constexpr int CP = C + 2;                       // LDS pad: kill bank conflicts
    __shared__ float sW[O * CP];
    __shared__ float sS[8][O];
    __shared__ float sQ[8][O];
    __shared__ float sA[O];
    __shared__ float sB[O];
    const int tid  = threadIdx.x;
    const int lane = tid & 31;
    const int wv   = tid >> 5;
    const int lr   = lane & 15;                     // n / m-within-tile
    const int lh   = lane >> 4;                     // lane half
    for (int i = tid; i < O * C; i += 256) sW[(i / C) * CP + (i % C)] = W[i];
    if (MODE == 2) for (int i = tid; i < O; i += 256) { sA[i] = As[i]; sB[i] = Bs[i]; }
    __syncthreads();

    const int m0 = blockIdx.x * 128 + wv * 16;
    v8f acc[OT];
    #pragma unroll
    for (int ot = 0; ot < OT; ++ot) {
        #pragma unroll
        for (int v = 0; v < 8; ++v) acc[ot][v] = bias[ot * 16 + v + 8 * lh];
    }
    const float* Fcol = F + m0 + lr;
    for (int kk = 0; kk < C; kk += 4) {
        const int krow = kk + lh * 2;               // B: k = lh*2 + v, n = lr
        v2f Bt;
        Bt.x = Fcol[(size_t)(krow    ) * M_TOT];
        Bt.y = Fcol[(size_t)(krow + 1) * M_TOT];
        #pragma unroll
        for (int ot = 0; ot < OT; ++ot) {
            const float* wp = &sW[(ot * 16 + lr) * CP + krow];  // A: m=lr, k=lh*2+v
            v2f At; At.x = wp[0]; At.y = wp[1];
            acc[ot] = __builtin_amdgcn_wmma_f32_16x16x4_f32(
                false, At, false, Bt, (short)0, acc[ot], false, false);
        }
    }

    if (MODE == 0 || MODE == 1) {
        #pragma unroll
        for (int ot = 0; ot < OT; ++ot) {
            #pragma unroll
            for (int v = 0; v < 8; ++v) {
                float s = acc[ot][v];                            // D: o=v+8*lh, n=lr
                if (MODE == 0)
                    Yout[(size_t)(ot * 16 + v + 8 * lh) * M_TOT + m0 + lr] = s;
                float q = s * s;
                #pragma unroll
                for (int msk = 1; msk < 16; msk <<= 1) {
                    s += __shfl_xor(s, msk, 32);
                    q += __shfl_xor(q, msk, 32);
                }
                if (lr == 0) { int o = ot * 16 + v + 8 * lh; sS[wv][o] = s; sQ[wv][o] = q; }
            }
        }
        __syncthreads();
        for (int o = tid; o < O; o += 256) {
            float s = 0.f, q = 0.f;
            #pragma unroll
            for (int w8 = 0; w8 < 8; ++w8) { s += sS[w8][o]; q += sQ[w8][o]; }
            psum[(size_t)o * NB + blockIdx.x] = s;
            psq [(size_t)o * NB + blockIdx.x] = q;
        }
    } else {
        #pragma unroll
        for (int ot = 0; ot < OT; ++ot) {
            #pragma unroll
            for (int v = 0; v < 8; ++v) {
                const int o = ot * 16 + v + 8 * lh;
                float val = fmaxf(sA[o] * acc[ot][v] + sB[o], 0.f);
                #pragma unroll
                for (int msk = 1; msk < 16; msk <<= 1)
                    val = fmaxf(val, __shfl_xor(val, msk, 32));  // max over 16 cols
                if (lr == 0) sS[wv][o] = val;
            }
        }
        __syncthreads();
        for (int i = tid; i < 4 * O; i += 256) {                 // 4 points per block
            const int pt = i / O;
            const int o  = i - pt * O;
            const float v = fmaxf(sS[2 * pt][o], sS[2 * pt + 1][o]);
            const int p = blockIdx.x * 4 + pt;
            const int b = p >> 12, n = p & (NPTS - 1);
            out[(size_t)OUT_PTS + ((size_t)b * O + o) * NPTS + n] = v;
        }
    }
}

// -------------------------------------------------------------------------
extern "C" void kernel_launch(void* const* d_in, const int* in_sizes, int n_in,
                              void* d_out, int out_size, void* d_ws, size_t ws_size,
                              hipStream_t stream) {
    const float* pts = (const float*)d_in[0];
    // d_in[1] = features (unused by reference)
    const float* W0 = (const float*)d_in[2];  const float* b0 = (const float*)d_in[3];
    const float* g0 = (const float*)d_in[4];  const float* be0 = (const float*)d_in[5];
    const float* W1 = (const float*)d_in[6];  const float* b1 = (const float*)d_in[7];
    const float* g1 = (const float*)d_in[8];  const float* be1 = (const float*)d_in[9];
    const float* W2 = (const float*)d_in[10]; const float* b2 = (const float*)d_in[11];
    const float* g2 = (const float*)d_in[12]; const float* be2 = (const float*)d_in[13];
    float* out = (float*)d_out;

    // workspace layout (floats):
    float* f1   = (float*)d_ws;                 // 64  x 524288  (y1 -> f1 in place)
    float* f2   = f1 + (size_t)64 * M_TOT;      // 128 x 524288  (y2 -> f2 in place)
    int*   idx  = (int*)(f2 + (size_t)128 * M_TOT); // 524288 ints
    float* psum = (float*)(idx + M_TOT);        // <= 256*8192
    float* psq  = psum + (size_t)256 * 8192;    // <= 256*8192
    float* As   = psq  + (size_t)256 * 8192;    // 256
    float* Bs   = As + 256;                     // 256

    const int NBg = M_TOT / 128;                // 4096 GEMM blocks
    const int NB1 = 256;                        // layer-1 stats partial blocks

    copy_points_kernel<<<(OUT_PTS + 255) / 256, 256, 0, stream>>>(pts, out);
    knn_kernel<<<dim3(NPTS / 256, BATCH), 256, 0, stream>>>(pts, idx);
    layer1_kernel<<<M_TOT / 256, 256, 0, stream>>>(pts, idx, W0, b0, f1);
    colstats_kernel<<<dim3(NB1, 64), 256, 0, stream>>>(f1, psum, psq, NB1);
    finalize_kernel<<<64, 256, 0, stream>>>(psum, psq, g0, be0, As, Bs, NB1);
    normrelu_kernel<<<dim3(M_TOT / 1024, 64), 256, 0, stream>>>(f1, As, Bs);

    gemm_wmma_kernel<64, 128, 0><<<NBg, 256, 0, stream>>>(
        f1, W1, b1, f2, psum, psq, nullptr, nullptr, nullptr, NBg);
    finalize_kernel<<<128, 256, 0, stream>>>(psum, psq, g1, be1, As, Bs, NBg);
    normrelu_kernel<<<dim3(M_TOT / 1024, 128), 256, 0, stream>>>(f2, As, Bs);

    gemm_wmma_kernel<128, 256, 1><<<NBg, 256, 0, stream>>>(
        f2, W2, b2, nullptr, psum, psq, nullptr, nullptr, nullptr, NBg);
    finalize_kernel<<<256, 256, 0, stream>>>(psum, psq, g2, be2, As, Bs, NBg);
    gemm_wmma_kernel<128, 256, 2><<<NBg, 256, 0, stream>>>(
        f2, W2, b2, nullptr, nullptr, nullptr, As, Bs, out, NBg);
}